// AuxIVA_T_ISS_46222438039634
// MI455X (gfx1250) — compile-verified
//
#include <hip/hip_runtime.h>
#include <math.h>

// ---------------------------------------------------------------------------
// AuxIVA-T-ISS for MI455X (gfx1250). B=4 batches, C=4 channels, F=257 freqs,
// N=1024 frames, 3 epochs, 2 taps, delay 1.
// L2-resident workload (Xc + X ~ 67MB << 192MB L2). Memory-path optimized:
// wave32 shuffle reductions, b128 vector traffic, async global->LDS staging
// for the per-(b,c,f) frame-axis reductions.
// ---------------------------------------------------------------------------

#define BB 4
#define CC 4
#define FF 257
#define NF 1024
#define NTAPS 2
#define PADF 3            /* N_TAPS + N_DELAY */
#define EPSM 1e-5f
#define EPSV 1e-3f
#define TPB 256
#define NWAVES (TPB/32)
#define M_ELEMS (BB*CC*FF*NF)

#ifndef __has_builtin
#define __has_builtin(x) 0
#endif

#if __has_builtin(__builtin_amdgcn_global_load_async_to_lds_b128)
#define ATH_ASYNC 1
#else
#define ATH_ASYNC 0
#endif

__device__ __forceinline__ void ath_wait_async() {
#if __has_builtin(__builtin_amdgcn_s_wait_asynccnt)
  __builtin_amdgcn_s_wait_asynccnt(0);
#else
  asm volatile("s_wait_asynccnt 0" ::: "memory");
#endif
}

#if ATH_ASYNC
// Builtin signature (from hipcc diagnostic): (v4i AS1*, v4i AS3*, imm, imm)
typedef int ath_v4i __attribute__((vector_size(4 * sizeof(int))));
typedef ath_v4i __attribute__((address_space(1)))* ath_gptr;
typedef ath_v4i __attribute__((address_space(3)))* ath_lptr;

__device__ __forceinline__ void ath_async_b128(const void* g, void* l) {
  __builtin_amdgcn_global_load_async_to_lds_b128((ath_gptr)g, (ath_lptr)l, 0, 0);
}
#endif

// ---- wave32 register-only reduction ----------------------------------------
__device__ __forceinline__ float wave_sum(float x) {
  #pragma unroll
  for (int off = 16; off > 0; off >>= 1) x += __shfl_down(x, off, 32);
  return x;
}

// ---- complex helpers (float2 = interleaved complex64) ----------------------
__device__ __forceinline__ float2 cmul(float2 a, float2 b) {
  float2 r; r.x = a.x*b.x - a.y*b.y; r.y = a.x*b.y + a.y*b.x; return r;
}
__device__ __forceinline__ float2 csub(float2 a, float2 b) {
  float2 r; r.x = a.x - b.x; r.y = a.y - b.y; return r;
}
__device__ __forceinline__ float2 cdiv(float2 a, float2 b) {
  float inv = 1.0f / (b.x*b.x + b.y*b.y);
  float2 r; r.x = (a.x*b.x + a.y*b.y)*inv; r.y = (a.y*b.x - a.x*b.y)*inv; return r;
}

// ---- ISS type-1 reduction: v[b,c,f]  (FIRST in file so disasm shows it) ----
__global__ void __launch_bounds__(TPB)
k_reduce1(const float2* __restrict__ Xc, const float* __restrict__ wts,
          float2* __restrict__ v, int src) {
  __shared__ __align__(16) float2 s_a[NF];           // Xc[b,c,f,:]
  __shared__ __align__(16) float2 s_s[NF];           // Xc[b,src,f,:]
  __shared__ __align__(16) float  s_w[NF];           // w[b,c,:]
  __shared__ float pr[NWAVES], pi[NWAVES], pd[NWAVES];
  int blk = blockIdx.x;                              // BB*CC*FF
  int f = blk % FF;
  int c = (blk / FF) % CC;
  int b = blk / (FF*CC);
  int tid = threadIdx.x;
  const float2* xc_row = Xc + (size_t)((b*CC + c)*FF + f)*NF;
  const float2* xs_row = Xc + (size_t)((b*CC + src)*FF + f)*NF;
  const float*  w_row  = wts + (b*CC + c)*NF;
#if ATH_ASYNC
  #pragma unroll
  for (int k = 0; k < (NF/2)/TPB; ++k) {
    int i = tid + k*TPB;
    ath_async_b128(xc_row + 2*i, &s_a[2*i]);         // 16B = 2 complex
    ath_async_b128(xs_row + 2*i, &s_s[2*i]);
  }
  ath_async_b128(w_row + 4*tid, &s_w[4*tid]);        // 16B = 4 floats
  ath_wait_async();
#else
  for (int i = tid; i < NF; i += TPB) { s_a[i] = xc_row[i]; s_s[i] = xs_row[i]; s_w[i] = w_row[i]; }
#endif
  __syncthreads();
  float nr = 0.f, ni = 0.f, dd = 0.f;
  #pragma unroll
  for (int k = 0; k < NF/TPB; ++k) {
    int i = tid + k*TPB;
    float2 a = s_a[i], s = s_s[i];
    float wn = s_w[i];
    nr = fmaf(wn, a.x*s.x + a.y*s.y, nr);            // a * conj(s)
    ni = fmaf(wn, a.y*s.x - a.x*s.y, ni);
    dd = fmaf(wn, s.x*s.x + s.y*s.y, dd);
  }
  nr = wave_sum(nr); ni = wave_sum(ni); dd = wave_sum(dd);
  int lane = tid & 31, wv = tid >> 5;
  if (lane == 0) { pr[wv] = nr; pi[wv] = ni; pd[wv] = dd; }
  __syncthreads();
  if (tid == 0) {
    float SR = 0.f, SI = 0.f, SD = 0.f;
    #pragma unroll
    for (int w8 = 0; w8 < NWAVES; ++w8) { SR += pr[w8]; SI += pi[w8]; SD += pd[w8]; }
    float invN = 1.0f / (float)NF;
    float den = fmaxf(SD*invN, EPSV);                // mean
    float2 out;
    if (c == src) { out.x = 1.0f - 1.0f/sqrtf(den); out.y = 0.0f; }
    else          { out.x = SR*invN/den;            out.y = SI*invN/den; }
    v[blk] = out;
  }
}

// ---- ISS type-2 reduction against delayed tap of ORIGINAL X ----------------
__global__ void __launch_bounds__(TPB)
k_reduce2(const float2* __restrict__ Xc, const float* __restrict__ Xr,
          const float* __restrict__ Xi, const float* __restrict__ wts,
          float2* __restrict__ v, int src, int tap) {
  __shared__ __align__(16) float2 s_a[NF];
  __shared__ __align__(16) float2 s_s[NF];
  __shared__ __align__(16) float  s_w[NF];
  __shared__ float pr[NWAVES], pi[NWAVES], pd[NWAVES];
  int blk = blockIdx.x;
  int f = blk % FF;
  int c = (blk / FF) % CC;
  int b = blk / (FF*CC);
  int tid = threadIdx.x;
  const float2* xc_row = Xc + (size_t)((b*CC + c)*FF + f)*NF;
  const float*  w_row  = wts + (b*CC + c)*NF;
  size_t xrow = (size_t)((b*CC + src)*FF + f)*NF;
  int sh = PADF - tap;                               // X_bar[t][n] = X[n+t-3]
#if ATH_ASYNC
  #pragma unroll
  for (int k = 0; k < (NF/2)/TPB; ++k) {
    int i = tid + k*TPB;
    ath_async_b128(xc_row + 2*i, &s_a[2*i]);
  }
  ath_async_b128(w_row + 4*tid, &s_w[4*tid]);
#else
  for (int i = tid; i < NF; i += TPB) { s_a[i] = xc_row[i]; s_w[i] = w_row[i]; }
#endif
  for (int i = tid; i < NF; i += TPB) {              // overlaps with async
    int j = i - sh;
    float2 s; s.x = 0.0f; s.y = 0.0f;
    if (j >= 0) { s.x = Xr[xrow + j]; s.y = Xi[xrow + j]; }
    s_s[i] = s;
  }
#if ATH_ASYNC
  ath_wait_async();
#endif
  __syncthreads();
  float nr = 0.f, ni = 0.f, dd = 0.f;
  #pragma unroll
  for (int k = 0; k < NF/TPB; ++k) {
    int i = tid + k*TPB;
    float2 a = s_a[i], s = s_s[i];
    float wn = s_w[i];
    nr = fmaf(wn, a.x*s.x + a.y*s.y, nr);
    ni = fmaf(wn, a.y*s.x - a.x*s.y, ni);
    dd = fmaf(wn, s.x*s.x + s.y*s.y, dd);
  }
  nr = wave_sum(nr); ni = wave_sum(ni); dd = wave_sum(dd);
  int lane = tid & 31, wv = tid >> 5;
  if (lane == 0) { pr[wv] = nr; pi[wv] = ni; pd[wv] = dd; }
  __syncthreads();
  if (tid == 0) {
    float SR = 0.f, SI = 0.f, SD = 0.f;
    #pragma unroll
    for (int w8 = 0; w8 < NWAVES; ++w8) { SR += pr[w8]; SI += pi[w8]; SD += pd[w8]; }
    float invN = 1.0f / (float)NF;
    float den = fmaxf(SD, EPSV);                     // denom is a SUM here
    float2 out; out.x = SR*invN/den; out.y = SI*invN/den;
    v[blk] = out;
  }
}

// ---- init: pack planar X_real/X_imag into interleaved Xc (in d_out) --------
__global__ void k_pack(const float* __restrict__ Xr, const float* __restrict__ Xi,
                       float2* __restrict__ Xc) {
  int gid = blockIdx.x*TPB + threadIdx.x;
  float2 v; v.x = Xr[gid]; v.y = Xi[gid];
  Xc[gid] = v;
}

// ---- init: W[b][i][f][j] = delta(i,j) --------------------------------------
__global__ void k_init_w(float2* __restrict__ W) {
  int gid = blockIdx.x*TPB + threadIdx.x;            // BB*CC*FF*CC
  int j = gid % CC;
  int i = (gid / (CC*FF)) % CC;
  float2 v; v.x = (i == j) ? 1.0f : 0.0f; v.y = 0.0f;
  W[gid] = v;
}

// ---- sumF[b,c,n] = sum_f |X|^2 (invariant up to per-(b,c) scale) -----------
__global__ void k_sumF(const float* __restrict__ Xr, const float* __restrict__ Xi,
                       float* __restrict__ sumF) {
  int gid = blockIdx.x*TPB + threadIdx.x;            // BB*CC*NF
  int n = gid % NF, bc = gid / NF;
  size_t base = (size_t)bc*FF*NF + n;
  float acc = 0.0f;
  for (int f = 0; f < FF; ++f) {
    float r = Xr[base + (size_t)f*NF], im = Xi[base + (size_t)f*NF];
    acc = fmaf(r, r, acc);
    acc = fmaf(im, im, acc);
  }
  sumF[gid] = acc;
}

// ---- meanv[b,c] = mean_{f,n} |X|^2 -----------------------------------------
__global__ void k_mean(const float* __restrict__ sumF, float* __restrict__ meanv) {
  __shared__ float pw[NWAVES];
  int bc = blockIdx.x, tid = threadIdx.x;
  const float* p = sumF + bc*NF;
  float acc = p[tid] + p[tid + 256] + p[tid + 512] + p[tid + 768];
  acc = wave_sum(acc);
  int lane = tid & 31, wv = tid >> 5;
  if (lane == 0) pw[wv] = acc;
  __syncthreads();
  if (tid == 0) {
    float S = 0.f;
    #pragma unroll
    for (int w8 = 0; w8 < NWAVES; ++w8) S += pw[w8];
    meanv[bc] = S / (float)(FF*NF);
  }
}

// ---- weights for a given epoch (Xloc rescale tracked analytically) ---------
__global__ void k_weights(const float* __restrict__ sumF, const float* __restrict__ meanv,
                          float* __restrict__ wts, int epoch) {
  int gid = blockIdx.x*TPB + threadIdx.x;            // BB*CC*NF
  int bc = gid / NF;
  float bm = meanv[bc];
  float scale2 = 1.0f, g = 1.0f;
  for (int e = 0; e <= epoch; ++e) {                 // |Xloc|^2 = scale2*|X|^2
    g = fmaxf(scale2*bm, EPSM);
    if (e < epoch) scale2 /= g;                      // Xloc /= sqrt(g)
  }
  float denom = 2.0f * sqrtf(scale2 * sumF[gid]);
  wts[gid] = g / fmaxf(denom, EPSM);                 // weights * g
}

// ---- ISS type-1 update: Xc[b,c,f,n] -= v[b,c,f]*Xc[b,src,f,n] --------------
__global__ void k_update1(float2* __restrict__ Xc, const float2* __restrict__ v, int src) {
  int gid = blockIdx.x*TPB + threadIdx.x;            // BB*FF*NF
  int n = gid % NF;
  int f = (gid / NF) % FF;
  int b = gid / (NF*FF);
  size_t base = ((size_t)(b*CC)*FF + f)*NF + n;
  float2 xs = Xc[base + (size_t)src*FF*NF];          // old Xc[src] held locally
  #pragma unroll
  for (int c = 0; c < CC; ++c) {
    float2 vc = v[(b*CC + c)*FF + f];
    size_t idx = base + (size_t)c*FF*NF;
    float2 x = Xc[idx];
    x.x -= vc.x*xs.x - vc.y*xs.y;
    x.y -= vc.x*xs.y + vc.y*xs.x;
    Xc[idx] = x;
  }
}

// ---- W update: W[b,i,f,j] -= v[b,i,f]*W[b,src,f,j] -------------------------
__global__ void k_updW(float2* __restrict__ W, const float2* __restrict__ v, int src) {
  int gid = blockIdx.x*TPB + threadIdx.x;            // BB*FF*CC (guarded)
  if (gid >= BB*FF*CC) return;
  int j = gid % CC;
  int f = (gid / CC) % FF;
  int b = gid / (CC*FF);
  float2 wsrc = W[((b*CC + src)*FF + f)*CC + j];
  #pragma unroll
  for (int i = 0; i < CC; ++i) {
    float2 vc = v[(b*CC + i)*FF + f];
    int idx = ((b*CC + i)*FF + f)*CC + j;
    float2 x = W[idx];
    x.x -= vc.x*wsrc.x - vc.y*wsrc.y;
    x.y -= vc.x*wsrc.y + vc.y*wsrc.x;
    W[idx] = x;
  }
}

// ---- ISS type-2 update: Xc[b,c,f,n] -= v[b,c,f]*Xbar[b,src,f,tap,n] --------
__global__ void k_update2(float2* __restrict__ Xc, const float* __restrict__ Xr,
                          const float* __restrict__ Xi, const float2* __restrict__ v,
                          int src, int tap) {
  int gid = blockIdx.x*TPB + threadIdx.x;            // BB*FF*NF
  int n = gid % NF;
  int f = (gid / NF) % FF;
  int b = gid / (NF*FF);
  int sh = PADF - tap;
  float2 xs; xs.x = 0.0f; xs.y = 0.0f;
  int j = n - sh;
  if (j >= 0) {
    size_t sidx = (size_t)((b*CC + src)*FF + f)*NF + j;
    xs.x = Xr[sidx]; xs.y = Xi[sidx];
  }
  size_t base = ((size_t)(b*CC)*FF + f)*NF + n;
  #pragma unroll
  for (int c = 0; c < CC; ++c) {
    float2 vc = v[(b*CC + c)*FF + f];
    size_t idx = base + (size_t)c*FF*NF;
    float2 x = Xc[idx];
    x.x -= vc.x*xs.x - vc.y*xs.y;
    x.y -= vc.x*xs.y + vc.y*xs.x;
    Xc[idx] = x;
  }
}

// ---- 4x4 complex solve per (b,f): WT a = e1, partial pivoting --------------
__global__ void k_solve(const float2* __restrict__ W, float2* __restrict__ A) {
  int gid = blockIdx.x*TPB + threadIdx.x;            // BB*FF (guarded)
  if (gid >= BB*FF) return;
  int f = gid % FF, b = gid / FF;
  float2 Mx[4][5];
  for (int j = 0; j < 4; ++j) {
    for (int i = 0; i < 4; ++i)
      Mx[j][i] = W[((b*CC + i)*FF + f)*CC + j];      // WT[j][i] = W[b,i,f,j]
    Mx[j][4].x = (j == 0) ? 1.0f : 0.0f; Mx[j][4].y = 0.0f;
  }
  for (int k = 0; k < 4; ++k) {
    int p = k;
    float best = Mx[k][k].x*Mx[k][k].x + Mx[k][k].y*Mx[k][k].y;
    for (int r = k + 1; r < 4; ++r) {
      float m = Mx[r][k].x*Mx[r][k].x + Mx[r][k].y*Mx[r][k].y;
      if (m > best) { best = m; p = r; }
    }
    if (p != k)
      for (int q = k; q < 5; ++q) { float2 t = Mx[k][q]; Mx[k][q] = Mx[p][q]; Mx[p][q] = t; }
    for (int r = k + 1; r < 4; ++r) {
      float2 fac = cdiv(Mx[r][k], Mx[k][k]);
      for (int q = k; q < 5; ++q) Mx[r][q] = csub(Mx[r][q], cmul(fac, Mx[k][q]));
    }
  }
  float2 x[4];
  for (int k = 3; k >= 0; --k) {
    float2 acc = Mx[k][4];
    for (int q = k + 1; q < 4; ++q) acc = csub(acc, cmul(Mx[k][q], x[q]));
    x[k] = cdiv(acc, Mx[k][k]);
  }
  for (int i = 0; i < 4; ++i) A[(b*FF + f)*CC + i] = x[i];
}

// ---- final: out = Xc * a (broadcast over frames), in place in d_out --------
__global__ void k_scale(float2* __restrict__ Xc, const float2* __restrict__ A) {
  int gid = blockIdx.x*TPB + threadIdx.x;
  int r = gid / NF;
  int f = r % FF;
  int c = (r / FF) % CC;
  int b = r / (FF*CC);
  float2 a = A[(b*FF + f)*CC + c];
  float2 x = Xc[gid];
  float2 o; o.x = x.x*a.x - x.y*a.y; o.y = x.x*a.y + x.y*a.x;
  Xc[gid] = o;
}

// ---------------------------------------------------------------------------
extern "C" void kernel_launch(void* const* d_in, const int* in_sizes, int n_in,
                              void* d_out, int out_size, void* d_ws, size_t ws_size,
                              hipStream_t stream) {
  (void)in_sizes; (void)n_in; (void)out_size; (void)ws_size;
  const float* Xr = (const float*)d_in[0];
  const float* Xi = (const float*)d_in[1];
  float2* Xc = (float2*)d_out;                       // Xc lives in d_out

  char* ws = (char*)d_ws;                            // ~329 KB total
  float2* W     = (float2*)ws;  ws += sizeof(float2)*BB*CC*FF*CC;
  float2* v     = (float2*)ws;  ws += sizeof(float2)*BB*CC*FF;
  float*  wts   = (float*)ws;   ws += sizeof(float)*BB*CC*NF;
  float*  sumF  = (float*)ws;   ws += sizeof(float)*BB*CC*NF;
  float*  meanv = (float*)ws;   ws += sizeof(float)*64;   // 16 used, keep 16B align
  float2* A     = (float2*)ws;

  k_pack  <<<M_ELEMS/TPB,        TPB, 0, stream>>>(Xr, Xi, Xc);
  k_init_w<<<(BB*CC*FF*CC)/TPB,  TPB, 0, stream>>>(W);
  k_sumF  <<<(BB*CC*NF)/TPB,     TPB, 0, stream>>>(Xr, Xi, sumF);
  k_mean  <<<BB*CC,              TPB, 0, stream>>>(sumF, meanv);

  for (int epoch = 0; epoch < 3; ++epoch) {
    k_weights<<<(BB*CC*NF)/TPB, TPB, 0, stream>>>(sumF, meanv, wts, epoch);
    for (int src = 0; src < CC; ++src) {
      k_reduce1<<<BB*CC*FF,                TPB, 0, stream>>>(Xc, wts, v, src);
      k_update1<<<(BB*FF*NF)/TPB,          TPB, 0, stream>>>(Xc, v, src);
      k_updW   <<<(BB*FF*CC + TPB-1)/TPB,  TPB, 0, stream>>>(W, v, src);
    }
    for (int src = 0; src < CC; ++src)
      for (int tap = 0; tap < NTAPS; ++tap) {
        k_reduce2<<<BB*CC*FF,       TPB, 0, stream>>>(Xc, Xr, Xi, wts, v, src, tap);
        k_update2<<<(BB*FF*NF)/TPB, TPB, 0, stream>>>(Xc, Xr, Xi, v, src, tap);
      }
  }

  k_solve<<<(BB*FF + TPB-1)/TPB, TPB, 0, stream>>>(W, A);
  k_scale<<<M_ELEMS/TPB,         TPB, 0, stream>>>(Xc, A);
}